// StructureLoss3D_42880953483929
// MI455X (gfx1250) — compile-verified
//
#include <hip/hip_runtime.h>
#include <cmath>

// ---------------------------------------------------------------------------
// StructureLoss3D for MI455X (gfx1250, wave32).
// All separable Gaussian convolutions are executed as banded-Toeplitz matmuls
// on V_WMMA_F32_16X16X4_F32 (16x16 output tile per wave, K accumulated in
// chunks of 4). Matrix log per voxel via cyclic Jacobi (registers only).
// ---------------------------------------------------------------------------

typedef float v2f __attribute__((ext_vector_type(2)));
typedef float v8f __attribute__((ext_vector_type(8)));

struct TapArg { float w[16]; };

#define BATCH 2
#define DIM   96
#define HW    (DIM*DIM)          // 9216
#define DHW   (DIM*DIM*DIM)      // 884736
#define NFE   (BATCH*DHW)        // 1769472 elements per field

// ---------------------------------------------------------------------------
// Generic 1-D convolution along an axis using WMMA f32 16x16x4.
//   out[u0+M, line N] = sum_t taps[t] * in[u0+M-R+t, line N]   (zero padded)
// Cast as D = A*B with A(M,K)=taps[K-M] (banded), B(K,N)=in[u0-R+K, line N].
// Axis parameterization: element index = base(line) + u*su,
//   base(l) = (l / L0)*S1 + (l % L0)*S0.   L0 is a multiple of 16.
// Grid: one 16(u) x 16(line) tile per wave; 8 waves (256 thr) per block.
// ---------------------------------------------------------------------------
template <int T, int R, int NC>
__global__ __launch_bounds__(256) void conv_wmma_kernel(
    const float* src, float* dst, TapArg taps,
    int U, int su, int L0, int S0, int S1, int uTiles)
{
    const int lane = threadIdx.x & 31;
    const int wv   = threadIdx.x >> 5;
    const int tile = blockIdx.x * 8 + wv;          // wave-uniform
    const int u0    = (tile % uTiles) * 16;
    const int lbase = (tile / uTiles) * 16;

    const int hi = lane >> 4;      // 0 or 1 (K offset +2 for upper half-wave)
    const int lo = lane & 15;      // M for A;  N (line) for B and D

    // Line handled by this lane (B column / D column).
    const int l    = lbase + lo;
    const int base = (l / L0) * S1 + (l % L0) * S0;

    // Precompute per-lane A values: A(M=lo, K=4c+2*hi+v) = taps[K - lo].
    float a0[NC], a1[NC];
#pragma unroll
    for (int c = 0; c < NC; ++c) {
        int K  = 4 * c + 2 * hi;
        int i0 = K - lo;
        int i1 = i0 + 1;
        int c0 = min(max(i0, 0), 15);
        int c1 = min(max(i1, 0), 15);
        float t0 = taps.w[c0];
        float t1 = taps.w[c1];
        a0[c] = (i0 >= 0 && i0 < T) ? t0 : 0.0f;
        a1[c] = (i1 >= 0 && i1 < T) ? t1 : 0.0f;
    }

    v8f acc = {0.f, 0.f, 0.f, 0.f, 0.f, 0.f, 0.f, 0.f};
#pragma unroll
    for (int c = 0; c < NC; ++c) {
        const int K   = 4 * c + 2 * hi;
        const int ua  = u0 - R + K;        // input u for B VGPR0
        const int ub  = ua + 1;            // input u for B VGPR1
        float b0 = (ua >= 0 && ua < U) ? src[base + ua * su] : 0.0f;
        float b1 = (ub >= 0 && ub < U) ? src[base + ub * su] : 0.0f;
        v2f A; A.x = a0[c]; A.y = a1[c];
        v2f B; B.x = b0;    B.y = b1;
        acc = __builtin_amdgcn_wmma_f32_16x16x4_f32(
            /*neg_a=*/false, A, /*neg_b=*/false, B,
            /*c_mod=*/(short)0, acc, /*reuse_a=*/false, /*reuse_b=*/false);
    }

    // D layout: VGPR i -> M = i + 8*hi, N = lo.
#pragma unroll
    for (int i = 0; i < 8; ++i) {
        int u = u0 + i + 8 * hi;
        dst[base + u * su] = acc[i];
    }
}

// ---------------------------------------------------------------------------
// Structure-tensor products (elementwise). Outputs may alias inputs:
// values are read into registers before any store.
// ---------------------------------------------------------------------------
__global__ __launch_bounds__(256) void products_kernel(
    const float* gx, const float* gy, const float* gz,
    float* c0, float* c1, float* c2, float* c3, float* c4, float* c5, int n)
{
    int i      = blockIdx.x * blockDim.x + threadIdx.x;
    int stride = gridDim.x * blockDim.x;
    for (; i < n; i += stride) {
        float x = gx[i], y = gy[i], z = gz[i];
        float xx = x * x, yy = y * y, zz = z * z;
        float xy = x * y, xz = x * z, yz = y * z;
        c0[i] = xx; c1[i] = yy; c2[i] = zz;
        c3[i] = xy; c4[i] = xz; c5[i] = yz;
    }
}

// ---------------------------------------------------------------------------
// 3x3 symmetric eigen-decomposition via cyclic Jacobi (constant indices only,
// fully register resident), then log reconstruction.
// ---------------------------------------------------------------------------
template <int P, int Q>
__device__ inline void jrot(float A[3][3], float V[3][3])
{
    float apq = A[P][Q];
    if (fabsf(apq) < 1e-28f) return;
    float theta = (A[Q][Q] - A[P][P]) / (2.0f * apq);
    float t = 1.0f / (fabsf(theta) + sqrtf(theta * theta + 1.0f));
    t = (theta >= 0.0f) ? t : -t;
    float c = 1.0f / sqrtf(t * t + 1.0f);
    float s = t * c;
#pragma unroll
    for (int k = 0; k < 3; ++k) {
        float akp = A[k][P], akq = A[k][Q];
        A[k][P] = c * akp - s * akq;
        A[k][Q] = s * akp + c * akq;
    }
#pragma unroll
    for (int k = 0; k < 3; ++k) {
        float apk = A[P][k], aqk = A[Q][k];
        A[P][k] = c * apk - s * aqk;
        A[Q][k] = s * apk + c * aqk;
    }
#pragma unroll
    for (int k = 0; k < 3; ++k) {
        float vkp = V[k][P], vkq = V[k][Q];
        V[k][P] = c * vkp - s * vkq;
        V[k][Q] = s * vkp + c * vkq;
    }
}

__device__ inline void logm_sym3(const float sv[6], float L[6])
{
    // sv = [xx, yy, zz, xy, xz, yz]
    float A[3][3] = {{sv[0], sv[3], sv[4]},
                     {sv[3], sv[1], sv[5]},
                     {sv[4], sv[5], sv[2]}};
    float V[3][3] = {{1.f, 0.f, 0.f}, {0.f, 1.f, 0.f}, {0.f, 0.f, 1.f}};
#pragma unroll
    for (int sw = 0; sw < 6; ++sw) {
        jrot<0, 1>(A, V);
        jrot<0, 2>(A, V);
        jrot<1, 2>(A, V);
    }
    float l0 = logf(fmaxf(A[0][0], 1e-6f));
    float l1 = logf(fmaxf(A[1][1], 1e-6f));
    float l2 = logf(fmaxf(A[2][2], 1e-6f));
    L[0] = V[0][0]*V[0][0]*l0 + V[0][1]*V[0][1]*l1 + V[0][2]*V[0][2]*l2;
    L[1] = V[1][0]*V[1][0]*l0 + V[1][1]*V[1][1]*l1 + V[1][2]*V[1][2]*l2;
    L[2] = V[2][0]*V[2][0]*l0 + V[2][1]*V[2][1]*l1 + V[2][2]*V[2][2]*l2;
    L[3] = V[0][0]*V[1][0]*l0 + V[0][1]*V[1][1]*l1 + V[0][2]*V[1][2]*l2;
    L[4] = V[0][0]*V[2][0]*l0 + V[0][1]*V[2][1]*l1 + V[0][2]*V[2][2]*l2;
    L[5] = V[1][0]*V[2][0]*l0 + V[1][1]*V[2][1]*l1 + V[1][2]*V[2][2]*l2;
}

__global__ void zero_kernel(float* out)
{
    if (threadIdx.x == 0 && blockIdx.x == 0) out[0] = 0.0f;
}

__global__ __launch_bounds__(256) void finalize_kernel(
    const float* fr, const float* ff, float* out, int n, float scale)
{
    float acc = 0.0f;
    for (int i = blockIdx.x * blockDim.x + threadIdx.x; i < n;
         i += gridDim.x * blockDim.x) {
        float sr[6], sf[6];
#pragma unroll
        for (int k = 0; k < 6; ++k) {
            sr[k] = fr[(size_t)i + (size_t)k * (size_t)n];
            sf[k] = ff[(size_t)i + (size_t)k * (size_t)n];
        }
        float Lr[6], Lf[6];
        logm_sym3(sr, Lr);
        logm_sym3(sf, Lf);
        float d0 = Lr[0] - Lf[0], d1 = Lr[1] - Lf[1], d2 = Lr[2] - Lf[2];
        float d3 = Lr[3] - Lf[3], d4 = Lr[4] - Lf[4], d5 = Lr[5] - Lf[5];
        acc += d0*d0 + d1*d1 + d2*d2 + 2.0f * (d3*d3 + d4*d4 + d5*d5);
    }
    __shared__ float red[256];
    red[threadIdx.x] = acc;
    __syncthreads();
    for (int s = 128; s > 0; s >>= 1) {
        if ((int)threadIdx.x < s) red[threadIdx.x] += red[threadIdx.x + s];
        __syncthreads();
    }
    if (threadIdx.x == 0) atomicAdd(out, red[0] * scale);
}

// ---------------------------------------------------------------------------
// Host orchestration. Workspace layout: 14 fields of NFE floats (~99 MB):
//   slots 0..5 : gradients / products / (fake blurred comps at the end)
//   slots 6..7 : blur ping-pong temps
//   slots 8..13: real blurred comps
// ---------------------------------------------------------------------------
extern "C" void kernel_launch(void* const* d_in, const int* in_sizes, int n_in,
                              void* d_out, int out_size, void* d_ws, size_t ws_size,
                              hipStream_t stream)
{
    const float* realv = (const float*)d_in[0];
    const float* fakev = (const float*)d_in[1];
    float* out = (float*)d_out;
    float* W   = (float*)d_ws;
    const size_t nfe = (size_t)NFE;
    auto F = [&](int s) { return W + (size_t)s * nfe; };

    // Gaussian taps (match reference: normalize g, then dg = (-t/sigma^2)*g).
    TapArg gS{}, dS{}, gR{};
    {
        double g[7], sum = 0.0;
        for (int i = 0; i < 7; ++i) { double x = i - 3; g[i] = std::exp(-0.5 * x * x); sum += g[i]; }
        for (int i = 0; i < 7; ++i) {
            g[i] /= sum;
            gS.w[i] = (float)g[i];
            dS.w[i] = (float)(-(double)(i - 3) * g[i]);   // sigma = 1
        }
    }
    {
        double g[13], sum = 0.0;
        for (int i = 0; i < 13; ++i) { double x = (i - 6) / 2.0; g[i] = std::exp(-0.5 * x * x); sum += g[i]; }
        for (int i = 0; i < 13; ++i) gR.w[i] = (float)(g[i] / sum);
    }

    // Axis parameterizations (see conv_wmma_kernel).
    struct Axis { int U, su, L0, S0, S1; };
    const Axis AX = {DIM, 1,   1 << 28, DIM, 0  };   // conv along W: base = line*96
    const Axis AY = {DIM, DIM, DIM,     1,   HW };   // conv along H
    const Axis AZ = {DIM, HW,  HW,      1,   DHW};   // conv along D (per batch)

    const int uTiles = DIM / 16;                 // 6
    const int tiles  = NFE / 256;                // 6912 tiles per field
    const int blocks = tiles / 8;                // 864 blocks of 8 waves

    auto convS = [&](const float* s, float* d, const TapArg& t, const Axis& a) {
        conv_wmma_kernel<7, 3, 6><<<blocks, 256, 0, stream>>>(
            s, d, t, a.U, a.su, a.L0, a.S0, a.S1, uTiles);
    };
    auto convR = [&](const float* s, float* d, const Axis& a) {
        conv_wmma_kernel<13, 6, 7><<<blocks, 256, 0, stream>>>(
            s, d, gR, a.U, a.su, a.L0, a.S0, a.S1, uTiles);
    };

    for (int inp = 0; inp < 2; ++inp) {
        const float* v = (inp == 0) ? realv : fakev;
        // Gradients (derivative-of-Gaussian, separable; 8 passes).
        convS(v,    F(0), gS, AX);   // a = Gx(v)
        convS(v,    F(1), dS, AX);   // b = Dx(v)
        convS(F(0), F(2), gS, AY);   // c = Gy(a)
        convS(F(0), F(3), dS, AY);   // d = Dy(a)
        convS(F(1), F(4), gS, AY);   // e = Gy(b)
        convS(F(2), F(5), dS, AZ);   // gz = Dz(c)
        convS(F(3), F(0), gS, AZ);   // gy = Gz(d)
        convS(F(4), F(1), gS, AZ);   // gx = Gz(e)
        // Products: gx=F1, gy=F0, gz=F5 -> comps in slots 0..5 (in-place safe).
        products_kernel<<<1024, 256, 0, stream>>>(
            F(1), F(0), F(5), F(0), F(1), F(2), F(3), F(4), F(5), (int)NFE);
        // Rho blur of each component (separable; ping-pong through slots 6,7).
        float* dest = (inp == 0) ? F(8) : F(0);
        for (int k = 0; k < 6; ++k) {
            convR(F(k), F(6), AX);
            convR(F(6), F(7), AY);
            convR(F(7), dest + (size_t)k * nfe, AZ);
        }
    }

    zero_kernel<<<1, 32, 0, stream>>>(out);
    finalize_kernel<<<2048, 256, 0, stream>>>(
        F(8), F(0), out, (int)NFE, 1.0f / (float)DHW);
}